// GCN_45406394253425
// MI455X (gfx1250) — compile-verified
//
#include <hip/hip_runtime.h>
#include <hip/hip_bf16.h>

typedef __attribute__((ext_vector_type(2))) float v2f;
typedef __attribute__((ext_vector_type(8))) float v8f;

#define DIN 128
#define DOUT 128

// ---------------- degree kernels ----------------
__global__ void k_init_deg(float* __restrict__ deg, int n) {
    int i = blockIdx.x * blockDim.x + threadIdx.x;
    if (i < n) deg[i] = 1.0f;  // self-loop
}

__global__ void k_accum_deg(float* __restrict__ deg, const long long* __restrict__ ei,
                            int e) {
    int i = blockIdx.x * blockDim.x + threadIdx.x;
    if (i < e) {
        long long d = ei[(size_t)e + i];  // dst row of edge_index
        __hip_atomic_fetch_add(deg + d, 1.0f, __ATOMIC_RELAXED, __HIP_MEMORY_SCOPE_AGENT);
    }
}

__global__ void k_rsqrt(float* __restrict__ deg, int n) {
    int i = blockIdx.x * blockDim.x + threadIdx.x;
    if (i < n) deg[i] = rsqrtf(deg[i]);  // in place: deg -> dinv
}

// ---------------- dense transform: h = x @ W  (WMMA f32 16x16x4) ----------------
// Block = 256 threads = 8 waves. Each wave computes one 16x16 output tile:
// rows [blockIdx.x*16, +16), cols [waveId*16, +16). K=128 in steps of 4.
// Epilogue: h -> workspace, out seeded with self-loop term h * dinv^2.
__global__ __launch_bounds__(256) void k_gemm_wmma(
    const float* __restrict__ x, const float* __restrict__ W,
    const float* __restrict__ dinv, float* __restrict__ h,
    float* __restrict__ out, int n) {
    const int wave = threadIdx.x >> 5;
    const int lane = threadIdx.x & 31;
    const int rowBase = blockIdx.x * 16;
    const int n0 = wave * 16;

    const int m = lane & 15;          // row within tile (A) / col within tile (B)
    const int koff = (lane >> 4) * 2; // 0 for lanes 0-15, 2 for lanes 16-31

    int rowL = rowBase + m;
    if (rowL > n - 1) rowL = n - 1;   // clamp loads, keep EXEC all-ones
    const float* xrow = x + (size_t)rowL * DIN;

    v8f c = {};
#pragma unroll
    for (int k0 = 0; k0 < DIN; k0 += 4) {
        v2f a, b;
        // A: 16x4 f32 tile. lane m holds A[m][k0+koff], A[m][k0+koff+1]
        a.x = xrow[k0 + koff];
        a.y = xrow[k0 + koff + 1];
        // B: 4x16 f32 tile. lane m holds B[k0+koff][n0+m], B[k0+koff+1][n0+m]
        b.x = W[(size_t)(k0 + koff) * DOUT + n0 + m];
        b.y = W[(size_t)(k0 + koff + 1) * DOUT + n0 + m];
        c = __builtin_amdgcn_wmma_f32_16x16x4_f32(
            /*neg_a=*/false, a, /*neg_b=*/false, b,
            /*c_mod=*/(short)0, c, /*reuse_a=*/false, /*reuse_b=*/false);
    }

    // C/D layout: VGPR r -> row M=r (lanes 0-15) / M=r+8 (lanes 16-31), col N=lane&15
    const int mrow = (lane >> 4) * 8;
#pragma unroll
    for (int r = 0; r < 8; ++r) {
        int row = rowBase + r + mrow;
        if (row < n) {
            int col = n0 + m;
            float v = c[r];
            size_t idx = (size_t)row * DOUT + col;
            h[idx] = v;
            float di = dinv[row];
            out[idx] = v * di * di;  // self-loop contribution, initializes out
        }
    }
}

// ---------------- edge aggregation: one wave32 per edge ----------------
// lane l handles 4 columns as a float4: gather h[src] (L2-resident), scale by
// dinv[src]*dinv[dst], scatter-add into out[dst] with native fp32 atomics.
__global__ __launch_bounds__(256) void k_agg_edges(
    const float* __restrict__ h, const float* __restrict__ dinv,
    const long long* __restrict__ ei, float* __restrict__ out, int e) {
    int wave = (int)((blockIdx.x * (size_t)blockDim.x + threadIdx.x) >> 5);
    int lane = threadIdx.x & 31;
    if (wave >= e) return;
    long long s = ei[wave];
    long long d = ei[(size_t)e + wave];
    float norm = dinv[s] * dinv[d];

    const float4* hs = (const float4*)(h + (size_t)s * DOUT);
    float4 v = hs[lane];
    float* po = out + (size_t)d * DOUT + lane * 4;
    __hip_atomic_fetch_add(po + 0, v.x * norm, __ATOMIC_RELAXED, __HIP_MEMORY_SCOPE_AGENT);
    __hip_atomic_fetch_add(po + 1, v.y * norm, __ATOMIC_RELAXED, __HIP_MEMORY_SCOPE_AGENT);
    __hip_atomic_fetch_add(po + 2, v.z * norm, __ATOMIC_RELAXED, __HIP_MEMORY_SCOPE_AGENT);
    __hip_atomic_fetch_add(po + 3, v.w * norm, __ATOMIC_RELAXED, __HIP_MEMORY_SCOPE_AGENT);
}

// ---------------- bias + ReLU, in place on out ----------------
__global__ void k_bias_relu(float* __restrict__ out, const float* __restrict__ bias,
                            int total4) {
    int i = blockIdx.x * blockDim.x + threadIdx.x;
    if (i >= total4) return;
    int col4 = i & (DOUT / 4 - 1);  // 32 float4 per row
    float4 b = ((const float4*)bias)[col4];
    float4 v = ((float4*)out)[i];
    v.x = fmaxf(v.x + b.x, 0.0f);
    v.y = fmaxf(v.y + b.y, 0.0f);
    v.z = fmaxf(v.z + b.z, 0.0f);
    v.w = fmaxf(v.w + b.w, 0.0f);
    ((float4*)out)[i] = v;
}

extern "C" void kernel_launch(void* const* d_in, const int* in_sizes, int n_in,
                              void* d_out, int out_size, void* d_ws, size_t ws_size,
                              hipStream_t stream) {
    const float* x = (const float*)d_in[0];
    const float* W = (const float*)d_in[1];
    const float* bias = (const float*)d_in[2];
    const long long* ei = (const long long*)d_in[3];
    float* out = (float*)d_out;

    const int n = in_sizes[0] / DIN;   // 100000 nodes
    const int e = in_sizes[3] / 2;     // 1600000 edges

    // workspace: [deg/dinv : n floats (padded)] [h : n*128 floats]
    float* deg = (float*)d_ws;
    size_t hOff = ((size_t)n + 255) & ~(size_t)255;
    float* h = deg + hOff;

    k_init_deg<<<(n + 255) / 256, 256, 0, stream>>>(deg, n);
    k_accum_deg<<<(e + 255) / 256, 256, 0, stream>>>(deg, ei, e);
    k_rsqrt<<<(n + 255) / 256, 256, 0, stream>>>(deg, n);
    k_gemm_wmma<<<(n + 15) / 16, 256, 0, stream>>>(x, W, deg, h, out, n);
    k_agg_edges<<<(e + 7) / 8, 256, 0, stream>>>(h, deg, ei, out, e);
    k_bias_relu<<<((n * (DOUT / 4)) + 255) / 256, 256, 0, stream>>>(out, bias, n * (DOUT / 4));
}